// GCNRecommender_1039382086189
// MI455X (gfx1250) — compile-verified
//
#include <hip/hip_runtime.h>

#define NN 50000
#define D  64
#define NTILES (NN / 16)   // 3125, exact

typedef float v2f __attribute__((ext_vector_type(2)));
typedef float v8f __attribute__((ext_vector_type(8)));

// ---------------- degree / normalization ----------------
__global__ __launch_bounds__(256) void k_deg_init(float* __restrict__ deg) {
    int n = blockIdx.x * 256 + threadIdx.x;
    if (n < NN) deg[n] = 1.0f;                       // self-loop
}

__global__ __launch_bounds__(256) void k_deg_count(const int* __restrict__ dst,
                                                   float* __restrict__ deg, int nE) {
    int e = blockIdx.x * 256 + threadIdx.x;
    if (e < nE) atomicAdd(&deg[dst[e]], 1.0f);
}

__global__ __launch_bounds__(256) void k_rsqrt(float* __restrict__ deg) {
    int n = blockIdx.x * 256 + threadIdx.x;
    if (n < NN) deg[n] = rsqrtf(deg[n]);             // deg >= 1 always
}

// ---------------- GEMM: hs = dis[row] * (in @ W), acc seeded = hs ----------
// One wave per 16-row tile; full-f32 WMMA 16x16x4; W transposed in LDS.
__global__ __launch_bounds__(256) void k_gemm_wmma(
    const float* __restrict__ in, const float* __restrict__ W,
    const float* __restrict__ dis, float* __restrict__ hs,
    float* __restrict__ acc)
{
    // W transposed into LDS: Wt[col][k], row stride 68 (8B-aligned, conflict-free reads)
    __shared__ float Wt[D * 68];
    for (int i = threadIdx.x; i < D * D; i += 256) {
        int k = i >> 6, c = i & 63;
        Wt[c * 68 + k] = W[i];
    }
    __syncthreads();

    int wave = (blockIdx.x * 256 + threadIdx.x) >> 5;
    if (wave >= NTILES) return;                      // whole-wave exit: EXEC stays all-ones
    int lane  = threadIdx.x & 31;
    int m     = lane & 15;                           // row within tile (A) / col within tile (B,C)
    int khalf = lane >> 4;                           // 0 or 1
    int row0  = wave * 16;
    int node  = row0 + m;

    // A tile: 16x64, per lane float2 per 4-wide K chunk: K = kc*4 + 2*khalf + {0,1}
    v2f A[16];
    const float* ap = in + node * D + 2 * khalf;
#pragma unroll
    for (int kc = 0; kc < 16; ++kc)
        A[kc] = *(const v2f*)(ap + kc * 4);

    // per-output-row scale d^{-1/2}; C layout: VGPR r holds M = r + 8*khalf
    float sc[8];
#pragma unroll
    for (int r = 0; r < 8; ++r) sc[r] = dis[row0 + khalf * 8 + r];

#pragma unroll
    for (int nt = 0; nt < 4; ++nt) {
        int col = nt * 16 + m;
        const float* wp = &Wt[col * 68 + 2 * khalf];
        v8f c = {};
#pragma unroll
        for (int kc = 0; kc < 16; ++kc) {
            v2f b = *(const v2f*)(wp + kc * 4);      // B[k][col], K = kc*4 + 2*khalf + {0,1}
            c = __builtin_amdgcn_wmma_f32_16x16x4_f32(
                    false, A[kc], false, b, (short)0, c, false, false);
        }
#pragma unroll
        for (int r = 0; r < 8; ++r) {
            int   orow = row0 + khalf * 8 + r;
            float v    = c[r] * sc[r];
            hs[orow * D + col]  = v;                 // message buffer
            acc[orow * D + col] = v;                 // seed = self-loop term
        }
    }
}

// ---------------- edge scatter: acc[dst] += hs[src] (pure atomic add) ------
__global__ __launch_bounds__(256) void k_scatter(
    const float* __restrict__ hs, const int* __restrict__ src,
    const int* __restrict__ dst, float* __restrict__ acc, int nE)
{
    int t = blockIdx.x * 256 + threadIdx.x;          // 16 threads per edge, float4 each
    int e = t >> 4;
    if (e >= nE) return;
    int c4 = (t & 15) << 2;
    int s = src[e], d = dst[e];
    float4 v = *(const float4*)(hs + s * D + c4);
    float* p = acc + d * D + c4;
    atomicAdd(p + 0, v.x);
    atomicAdd(p + 1, v.y);
    atomicAdd(p + 2, v.z);
    atomicAdd(p + 3, v.w);
}

// ---------------- finalize: out = [relu](dis[n]*acc + b) -------------------
template <bool RELU>
__global__ __launch_bounds__(256) void k_finalize(
    const float* __restrict__ acc, const float* __restrict__ dis,
    const float* __restrict__ b, float* __restrict__ out)
{
    int t = blockIdx.x * 256 + threadIdx.x;
    if (t >= NN * D) return;
    int n = t >> 6, c = t & 63;
    float v = dis[n] * acc[t] + b[c];
    if (RELU) v = fmaxf(v, 0.0f);
    out[t] = v;
}

// ---------------------------------------------------------------------------
extern "C" void kernel_launch(void* const* d_in, const int* in_sizes, int n_in,
                              void* d_out, int out_size, void* d_ws, size_t ws_size,
                              hipStream_t stream)
{
    (void)n_in; (void)out_size; (void)ws_size;
    const float* x  = (const float*)d_in[0];
    const int*   ei = (const int*)  d_in[1];
    const float* W0 = (const float*)d_in[2];
    const float* b0 = (const float*)d_in[3];
    const float* W1 = (const float*)d_in[4];
    const float* b1 = (const float*)d_in[5];
    const float* W2 = (const float*)d_in[6];
    const float* b2 = (const float*)d_in[7];
    float* out = (float*)d_out;

    int nE = in_sizes[1] / 2;
    const int* src = ei;
    const int* dst = ei + nE;

    float* ws  = (float*)d_ws;
    float* dis = ws;                 // 50000 (padded to 50048)
    float* hsb = ws + 50048;         // 3.2M floats
    float* acc = hsb + NN * D;       // 3.2M floats

    int nbN = (NN + 255) / 256;
    int nbE = (nE + 255) / 256;
    int nbG = (NTILES + 7) / 8;              // 8 waves/block
    int nbS = (nE * 16 + 255) / 256;
    int nbF = (NN * D + 255) / 256;

    k_deg_init <<<nbN, 256, 0, stream>>>(dis);
    k_deg_count<<<nbE, 256, 0, stream>>>(dst, dis, nE);
    k_rsqrt    <<<nbN, 256, 0, stream>>>(dis);

    // layer 0 (ReLU fused into finalize store)
    k_gemm_wmma      <<<nbG, 256, 0, stream>>>(x,   W0, dis, hsb, acc);
    k_scatter        <<<nbS, 256, 0, stream>>>(hsb, src, dst, acc, nE);
    k_finalize<true> <<<nbF, 256, 0, stream>>>(acc, dis, b0, hsb);
    // layer 1
    k_gemm_wmma      <<<nbG, 256, 0, stream>>>(hsb, W1, dis, hsb, acc);
    k_scatter        <<<nbS, 256, 0, stream>>>(hsb, src, dst, acc, nE);
    k_finalize<true> <<<nbF, 256, 0, stream>>>(acc, dis, b1, hsb);
    // layer 2 (no ReLU on network output)
    k_gemm_wmma      <<<nbG, 256, 0, stream>>>(hsb, W2, dis, hsb, acc);
    k_scatter        <<<nbS, 256, 0, stream>>>(hsb, src, dst, acc, nE);
    k_finalize<false><<<nbF, 256, 0, stream>>>(acc, dis, b2, out);
}